// ObliviousDecisionTree_55774445306400
// MI455X (gfx1250) — compile-verified
//
#include <hip/hip_runtime.h>

typedef __attribute__((ext_vector_type(2))) float v2f;
typedef __attribute__((ext_vector_type(8))) float v8f;

#define BATCH     262144
#define KDIM      256
#define DEPTH     10
#define NLEAF     1024
#define TILE_ROWS 256
#define NPAD      16   // N padded 10 -> 16 for the 16x16 WMMA tile

__global__ __launch_bounds__(256)
void odt_kernel(const float* __restrict__ x,
                const float* __restrict__ W,
                const float* __restrict__ b,
                const float* __restrict__ leaf,
                float* __restrict__ out)
{
    // LDS: leaf table (4KB) + WMMA-B-layout W (16KB) + z staging tile (16KB) + bias
    __shared__ float s_leaf[NLEAF];
    __shared__ float s_wp[(KDIM / 4) * 2 * NPAD * 2];  // [kk][half][n][kpair]
    __shared__ float s_z[TILE_ROWS * NPAD];
    __shared__ float s_b[NPAD];

    const int tid = threadIdx.x;

    // ---- stage leaf values ----
    for (int i = tid; i < NLEAF; i += 256) s_leaf[i] = leaf[i];
    // ---- stage bias (zero-pad N to 16) ----
    if (tid < NPAD) s_b[tid] = (tid < DEPTH) ? b[tid] : 0.0f;
    // ---- stage W into the WMMA B-operand lane layout ----
    // global k = kk*4 + half*2 + j ; lane n reads s_wp[((kk*2+half)*16 + n)*2 + {0,1}]
    for (int i = tid; i < (KDIM / 4) * 2 * NPAD * 2; i += 256) {
        int j  = i & 1;
        int n  = (i >> 1) & 15;
        int h  = (i >> 5) & 1;
        int kk = i >> 6;
        int k  = kk * 4 + h * 2 + j;
        s_wp[i] = (n < DEPTH) ? W[k * DEPTH + n] : 0.0f;
    }
    __syncthreads();

    // =================== Phase 1: z = x@W + b via V_WMMA_F32_16X16X4_F32 ===================
    const int wv   = tid >> 5;
    const int lane = tid & 31;
    const int half = lane >> 4;   // 0: lanes 0-15 (K 0,1) ; 1: lanes 16-31 (K 2,3)
    const int l16  = lane & 15;
    const long long rowBase = (long long)blockIdx.x * TILE_ROWS;

    for (int t = 0; t < 2; ++t) {                 // 8 waves x 2 tiles = 256 rows
        const int tile = wv * 2 + t;
        const long long arow = rowBase + tile * 16 + l16;   // A-matrix M = lane&15
        const float* xr = x + arow * KDIM + half * 2;

        v8f c;
        #pragma unroll
        for (int r = 0; r < 8; ++r) c[r] = s_b[l16];        // C lane holds N = lane&15

        #pragma unroll 8
        for (int kk = 0; kk < KDIM / 4; ++kk) {
            v2f a  = *(const v2f*)(xr + kk * 4);                              // x[row][k..k+1]
            v2f bm = *(const v2f*)(&s_wp[((kk * 2 + half) * NPAD + l16) * 2]); // W[k..k+1][n]
            c = __builtin_amdgcn_wmma_f32_16x16x4_f32(false, a, false, bm,
                                                      (short)0, c, false, false);
        }
        // C/D layout: VGPR r, lanes 0-15 -> M=r ; lanes 16-31 -> M=8+r ; N = lane&15
        #pragma unroll
        for (int r = 0; r < 8; ++r)
            s_z[(tile * 16 + r + 8 * half) * NPAD + l16] = c[r];
    }
    __syncthreads();

    // =================== Phase 2: sigmoid + oblivious-tree contraction ===================
    // one thread per batch row
    float dec[DEPTH];
    #pragma unroll
    for (int d = 0; d < DEPTH; ++d) {
        float z = s_z[tid * NPAD + d];
        dec[d] = 1.0f / (1.0f + __expf(-z));
    }

    // leaf index l = b0..b9 (b0 = MSB). Split l = (hi<<5)|lo :
    //   hi bits 4..0 = b0..b4, lo bits 4..0 = b5..b9.
    // w_hi = tensorprod over d=0..4 of [1-dec_d, dec_d] ; w_lo over d=5..9.
    float wh[32], wl[32];
    wh[0] = 1.0f;
    #pragma unroll
    for (int d = 0; d < 5; ++d) {
        const int sz = 1 << d;
        #pragma unroll
        for (int j2 = sz - 1; j2 >= 0; --j2) {
            float p = wh[j2];
            wh[2 * j2 + 1] = p * dec[d];
            wh[2 * j2]     = p * (1.0f - dec[d]);
        }
    }
    wl[0] = 1.0f;
    #pragma unroll
    for (int d = 5; d < 10; ++d) {
        const int sz = 1 << (d - 5);
        #pragma unroll
        for (int j2 = sz - 1; j2 >= 0; --j2) {
            float p = wl[j2];
            wl[2 * j2 + 1] = p * dec[d];
            wl[2 * j2]     = p * (1.0f - dec[d]);
        }
    }

    // out = sum_hi w_hi[hi] * sum_j leaf[hi*32+j] * w_lo[j]
    // leaf reads are wave-uniform -> LDS broadcast, conflict-free.
    float acc = 0.0f;
    #pragma unroll
    for (int hi = 0; hi < 32; ++hi) {
        float s = 0.0f;
        #pragma unroll
        for (int j = 0; j < 32; ++j)
            s = fmaf(s_leaf[hi * 32 + j], wl[j], s);
        acc = fmaf(wh[hi], s, acc);
    }

    out[rowBase + tid] = acc;
}

extern "C" void kernel_launch(void* const* d_in, const int* in_sizes, int n_in,
                              void* d_out, int out_size, void* d_ws, size_t ws_size,
                              hipStream_t stream) {
    const float* x    = (const float*)d_in[0];
    const float* W    = (const float*)d_in[1];
    const float* b    = (const float*)d_in[2];
    const float* leaf = (const float*)d_in[3];
    float* out = (float*)d_out;
    (void)in_sizes; (void)n_in; (void)out_size; (void)d_ws; (void)ws_size;

    dim3 grid(BATCH / TILE_ROWS);
    dim3 block(256);
    odt_kernel<<<grid, block, 0, stream>>>(x, W, b, leaf, out);
}